// remove_color_15822659518467
// MI455X (gfx1250) — compile-verified
//
#include <hip/hip_runtime.h>

// Cost-volume: out[b][s][y][x] = sum_c |img[b][c][y][x] - pad_inf(img)[b][c][y+dy][x+dx]|
// s = dy*7+dx, dy,dx in [0,7), pad radius 3, pad value +inf.
// B=16, C=3, H=W=384 -> out [16,49,384,384] f32.

#define RAD     3
#define KDIM    7            // 2*RAD+1
#define NSHIFT  49
#define TILE    32
#define HALO    (TILE + 2*RAD)   // 38
#define LSTRIDE 40               // padded LDS row stride in floats
#define CC      3
#define HH      384
#define WW      384
#define NTHREADS 256

typedef float v4f __attribute__((ext_vector_type(4)));
typedef float v2f __attribute__((ext_vector_type(2)));

__global__ __launch_bounds__(NTHREADS)
void cost_volume_l1_kernel(const float* __restrict__ img, float* __restrict__ out)
{
    __shared__ float tile[CC * HALO * LSTRIDE];   // 3*38*40*4 = 18,240 B

    const int tid = threadIdx.x;
    const int tx0 = blockIdx.x * TILE;
    const int ty0 = blockIdx.y * TILE;
    const int b   = blockIdx.z;

    // ---- 1) fill the whole LDS tile with +inf (the pad value) ----
    const float INF = __builtin_inff();
    for (int i = tid; i < CC * HALO * LSTRIDE; i += NTHREADS)
        tile[i] = INF;
    __syncthreads();   // compiler emits s_wait_dscnt + barrier

    // ---- 2) async global->LDS copy of the in-bounds halo tile (ASYNCcnt path) ----
    const float* imgb = img + (size_t)b * CC * HH * WW;
    for (int i = tid; i < CC * HALO * HALO; i += NTHREADS) {
        const int c  = i / (HALO * HALO);
        const int r_ = i - c * (HALO * HALO);
        const int y  = r_ / HALO;
        const int x  = r_ - y * HALO;
        const int gy = ty0 - RAD + y;
        const int gx = tx0 - RAD + x;
        if (gy >= 0 && gy < HH && gx >= 0 && gx < WW) {
            // LDS byte offset = low 32 bits of the generic shared-pointer address
            unsigned lds_addr = (unsigned)(size_t)&tile[(c * HALO + y) * LSTRIDE + x];
            unsigned long long gaddr =
                (unsigned long long)(size_t)(imgb + ((size_t)c * HH + gy) * WW + gx);
            asm volatile("global_load_async_to_lds_b32 %0, %1, off"
                         :: "v"(lds_addr), "v"(gaddr) : "memory");
        }
    }
    // each wave waits for its own async transfers, then workgroup barrier
    asm volatile("s_wait_asynccnt 0" ::: "memory");
    __syncthreads();

    // ---- 3) compute: each thread owns 4 consecutive x pixels of one row ----
    const int lx0 = (tid & 7) * 4;   // 0,4,...,28
    const int ly  = tid >> 3;        // 0..31
    const int gy  = ty0 + ly;
    const int gx0 = tx0 + lx0;

    // center values (always in-bounds), 3 channels x 4 pixels
    float ctr[CC][4];
    #pragma unroll
    for (int c = 0; c < CC; ++c) {
        #pragma unroll
        for (int px = 0; px < 4; ++px)
            ctr[c][px] = tile[(c * HALO + (ly + RAD)) * LSTRIDE + (lx0 + RAD + px)];
    }

    const size_t out_b = (size_t)b * NSHIFT * HH * WW;

    for (int dy = 0; dy < KDIM; ++dy) {
        // 10-wide sliding window per channel: cols lx0 .. lx0+9 of row ly+dy
        float r[CC][10];
        #pragma unroll
        for (int c = 0; c < CC; ++c) {
            const int base = (c * HALO + (ly + dy)) * LSTRIDE + lx0; // 16B aligned
            const v4f a = *(const v4f*)&tile[base];
            const v4f bq = *(const v4f*)&tile[base + 4];
            const v2f cq = *(const v2f*)&tile[base + 8];
            r[c][0] = a.x; r[c][1] = a.y; r[c][2] = a.z; r[c][3] = a.w;
            r[c][4] = bq.x; r[c][5] = bq.y; r[c][6] = bq.z; r[c][7] = bq.w;
            r[c][8] = cq.x; r[c][9] = cq.y;
        }
        #pragma unroll
        for (int dx = 0; dx < KDIM; ++dx) {
            v4f o;
            #pragma unroll
            for (int px = 0; px < 4; ++px) {
                float v = __builtin_fabsf(ctr[0][px] - r[0][dx + px]);
                v      += __builtin_fabsf(ctr[1][px] - r[1][dx + px]);
                v      += __builtin_fabsf(ctr[2][px] - r[2][dx + px]);
                o[px] = v;
            }
            const int s = dy * KDIM + dx;
            float* dst = out + out_b + (((size_t)s * HH + gy) * WW + gx0);
            __builtin_nontemporal_store(o, (v4f*)dst);   // 462 MB streamed once: NT
        }
    }
}

extern "C" void kernel_launch(void* const* d_in, const int* in_sizes, int n_in,
                              void* d_out, int out_size, void* d_ws, size_t ws_size,
                              hipStream_t stream)
{
    (void)in_sizes; (void)n_in; (void)d_ws; (void)ws_size; (void)out_size;
    const float* img = (const float*)d_in[0];
    float* out = (float*)d_out;

    dim3 grid(WW / TILE, HH / TILE, 16);   // 12 x 12 x 16
    dim3 block(NTHREADS);
    cost_volume_l1_kernel<<<grid, block, 0, stream>>>(img, out);
}